// XLMRCorefModel_84765474554682
// MI455X (gfx1250) — compile-verified
//
#include <hip/hip_runtime.h>
#include <math.h>

#define S_LEN 1024
#define HID   768
#define NL    50
#define SW    10
#define D3    2304   // 3*HID

typedef __attribute__((ext_vector_type(2))) float v2f;
typedef __attribute__((ext_vector_type(8))) float v8f;

// ---------------- small helpers ----------------
__device__ __forceinline__ float wred16(float v) {
    // reduce across the 16-lane half-wave (never crosses bit 4)
    v += __shfl_xor(v, 1); v += __shfl_xor(v, 2);
    v += __shfl_xor(v, 4); v += __shfl_xor(v, 8);
    return v;
}

// ---------------- GCN: counts / bias / degree ----------------
__global__ void cnt_count_kernel(const int* __restrict__ edges, float* __restrict__ cnt, int E) {
    int e = blockIdx.x * 256 + threadIdx.x;
    if (e >= E) return;
    int tgt = edges[3 * e + 1];
    int lbl = edges[3 * e + 2];
    atomicAdd(cnt + tgt * NL + lbl, 1.0f);
}

__global__ void deg_kernel(const float* __restrict__ cnt, float* __restrict__ deg) {
    int s = blockIdx.x * 256 + threadIdx.x;
    if (s >= S_LEN) return;
    float t = 0.f;
    for (int l = 0; l < NL; ++l) t += cnt[s * NL + l];
    deg[s] = t > 1.f ? t : 1.f;
}

__global__ void bias_init_kernel(const float* __restrict__ cnt, const float* __restrict__ gb,
                                 float* __restrict__ gout) {
    int t = blockIdx.x * 256 + threadIdx.x;
    if (t >= S_LEN * HID) return;
    int s = t / HID, d = t % HID;
    float a = 0.f;
    for (int l = 0; l < NL; ++l) a += cnt[s * NL + l] * gb[l * HID + d];
    gout[t] = a;
}

// per-label scatter (fallback path): x[src] -> agg[tgt]; 16 threads/edge, 48 floats each
__global__ void gcn_scatter_kernel(const int* __restrict__ edges, const float* __restrict__ x,
                                   float* __restrict__ agg, int label, int E) {
    int t = blockIdx.x * 256 + threadIdx.x;
    int e = t >> 4;
    int c = t & 15;
    if (e >= E) return;
    if (edges[3 * e + 2] != label) return;
    int src = edges[3 * e + 0];
    int tgt = edges[3 * e + 1];
    const float* xs = x + (size_t)src * HID + c * 48;
    float* ag = agg + (size_t)tgt * HID + c * 48;
    #pragma unroll 8
    for (int i = 0; i < 48; ++i) atomicAdd(ag + i, xs[i]);
}

// single-pass scatter (big path): x[src] -> agg[tgt*NL + lbl]
__global__ void gcn_scatter_all_kernel(const int* __restrict__ edges, const float* __restrict__ x,
                                       float* __restrict__ agg, int E) {
    int t = blockIdx.x * 256 + threadIdx.x;
    int e = t >> 4;
    int c = t & 15;
    if (e >= E) return;
    int src = edges[3 * e + 0];
    int tgt = edges[3 * e + 1];
    int lbl = edges[3 * e + 2];
    const float* xs = x + (size_t)src * HID + c * 48;
    float* ag = agg + ((size_t)tgt * NL + lbl) * HID + c * 48;
    #pragma unroll 8
    for (int i = 0; i < 48; ++i) atomicAdd(ag + i, xs[i]);
}

// ---------------- WMMA fp32 GEMM core: one wave -> 16x64 strip ----------------
__device__ __forceinline__ void wmma_k_loop(const float* __restrict__ A, int lda,
                                            const float* __restrict__ B, int ldb,
                                            int K, int row0, int col0,
                                            int m, int hi, v8f acc[4]) {
    for (int k = 0; k < K; k += 4) {
        // A 16x4 f32: lane = m + 16*hi holds A[m, 2*hi] / A[m, 2*hi+1] in VGPR0/1
        const float* ap = A + (size_t)(row0 + m) * lda + k + 2 * hi;
        v2f a; a.x = ap[0]; a.y = ap[1];
        const float* bp = B + (size_t)(k + 2 * hi) * ldb + col0 + m;
        #pragma unroll
        for (int t = 0; t < 4; ++t) {
            // B 4x16 f32: lane n=m holds B[2*hi, n] / B[2*hi+1, n] in VGPR0/1
            v2f b; b.x = bp[t * 16]; b.y = bp[t * 16 + ldb];
            acc[t] = __builtin_amdgcn_wmma_f32_16x16x4_f32(
                false, a, false, b, (short)0, acc[t], false, false);
        }
    }
}

// GCN GEMM (fallback): C(1024x768) += A(1024x768) @ B(768x768)   (C pre-initialized)
__global__ __launch_bounds__(128) void gemm_gcn_kernel(const float* __restrict__ A,
                                                       const float* __restrict__ B,
                                                       float* __restrict__ C) {
    const int lda = HID, ldb = HID, ldc = HID, K = HID;
    int wave = threadIdx.x >> 5;
    int lane = threadIdx.x & 31;
    int m = lane & 15, hi = lane >> 4;
    int row0 = blockIdx.x * 64 + wave * 16;
    int col0 = blockIdx.y * 64;
    v8f acc[4];
    #pragma unroll
    for (int t = 0; t < 4; ++t)
        #pragma unroll
        for (int g = 0; g < 8; ++g) acc[t][g] = 0.f;
    wmma_k_loop(A, lda, B, ldb, K, row0, col0, m, hi, acc);
    #pragma unroll
    for (int t = 0; t < 4; ++t)
        #pragma unroll
        for (int g = 0; g < 8; ++g)
            C[(size_t)(row0 + g + 8 * hi) * ldc + col0 + t * 16 + m] += acc[t][g];
}

// GCN GEMM (big path): K=38400 split over grid.z (one 768-chunk per label),
// atomic accumulate into bias-initialized C. AGG lda = NL*HID, gcn_W flat ldb = HID.
__global__ __launch_bounds__(128) void gemm_gcn_big_kernel(const float* __restrict__ A,
                                                           const float* __restrict__ B,
                                                           float* __restrict__ C) {
    const int lda = NL * HID, ldb = HID, ldc = HID, KC = HID;
    int wave = threadIdx.x >> 5;
    int lane = threadIdx.x & 31;
    int m = lane & 15, hi = lane >> 4;
    int row0 = blockIdx.x * 64 + wave * 16;
    int col0 = blockIdx.y * 64;
    int kbase = blockIdx.z * KC;
    v8f acc[4];
    #pragma unroll
    for (int t = 0; t < 4; ++t)
        #pragma unroll
        for (int g = 0; g < 8; ++g) acc[t][g] = 0.f;
    wmma_k_loop(A + kbase, lda, B + (size_t)kbase * ldb, ldb, KC, row0, col0, m, hi, acc);
    #pragma unroll
    for (int t = 0; t < 4; ++t)
        #pragma unroll
        for (int g = 0; g < 8; ++g)
            atomicAdd(&C[(size_t)(row0 + g + 8 * hi) * ldc + col0 + t * 16 + m], acc[t][g]);
}

// Mention GEMM, fused: msc[n] += sum_j relu( (SV@W1)[n,j] + b1[j] ) * w2[j]
__global__ __launch_bounds__(128) void gemm_mention_kernel(const float* __restrict__ A,
                                                           const float* __restrict__ W1,
                                                           const float* __restrict__ b1,
                                                           const float* __restrict__ w2,
                                                           float* __restrict__ msc,
                                                           int nspans) {
    const int lda = D3, ldb = D3, K = D3;
    int wave = threadIdx.x >> 5;
    int lane = threadIdx.x & 31;
    int m = lane & 15, hi = lane >> 4;
    int row0 = blockIdx.x * 64 + wave * 16;
    int col0 = blockIdx.y * 64;
    v8f acc[4];
    #pragma unroll
    for (int t = 0; t < 4; ++t)
        #pragma unroll
        for (int g = 0; g < 8; ++g) acc[t][g] = 0.f;
    wmma_k_loop(A, lda, W1, ldb, K, row0, col0, m, hi, acc);

    float rowsum[8];
    #pragma unroll
    for (int g = 0; g < 8; ++g) rowsum[g] = 0.f;
    #pragma unroll
    for (int t = 0; t < 4; ++t) {
        int j = col0 + t * 16 + m;
        float bj = b1[j], wj = w2[j];
        #pragma unroll
        for (int g = 0; g < 8; ++g) {
            float h = acc[t][g] + bj;
            h = h > 0.f ? h : 0.f;
            rowsum[g] += h * wj;
        }
    }
    #pragma unroll
    for (int g = 0; g < 8; ++g) {
        float v = wred16(rowsum[g]);
        if (m == 0) {
            int row = row0 + g + 8 * hi;
            if (row < nspans) atomicAdd(msc + row, v);
        }
    }
}

// ---------------- emb = x + relu(out/deg) ----------------
__global__ void emb_kernel(const float* __restrict__ x, const float* __restrict__ gout,
                           const float* __restrict__ deg, float* __restrict__ emb) {
    int t = blockIdx.x * 256 + threadIdx.x;
    if (t >= S_LEN * HID) return;
    int s = t / HID;
    float v = gout[t] / deg[s];
    v = v > 0.f ? v : 0.f;
    emb[t] = x[t] + v;
}

// ---------------- span attention + span_vec assembly (block=320 = 10 waves) ----------------
__global__ __launch_bounds__(320) void span_kernel(const float* __restrict__ emb,
                                                   const int* __restrict__ starts,
                                                   const int* __restrict__ ends,
                                                   const float* __restrict__ attn_w,
                                                   const float* __restrict__ attn_b,
                                                   float* __restrict__ span_vec) {
    int n = blockIdx.x;
    int start = starts[n], end = ends[n];
    int wave = threadIdx.x >> 5, lane = threadIdx.x & 31;
    __shared__ float ssc[SW];

    int pos = start + wave;
    int pc = pos < (S_LEN - 1) ? pos : (S_LEN - 1);
    float p = 0.f;
    const float* er = emb + (size_t)pc * HID;
    for (int h = lane; h < HID; h += 32) p += er[h] * attn_w[h];
    p += __shfl_xor(p, 1); p += __shfl_xor(p, 2); p += __shfl_xor(p, 4);
    p += __shfl_xor(p, 8); p += __shfl_xor(p, 16);
    if (lane == 0) ssc[wave] = (pos <= end) ? (p + attn_b[0]) : -1e9f;
    __syncthreads();

    float s[SW], mx = -1e30f;
    #pragma unroll
    for (int w = 0; w < SW; ++w) { s[w] = ssc[w]; mx = s[w] > mx ? s[w] : mx; }
    float sum = 0.f;
    #pragma unroll
    for (int w = 0; w < SW; ++w) { s[w] = expf(s[w] - mx); sum += s[w]; }
    float inv = 1.f / sum;

    float* sv = span_vec + (size_t)n * D3;
    const float* es = emb + (size_t)start * HID;
    const float* ee = emb + (size_t)end * HID;
    for (int h = threadIdx.x; h < HID; h += 320) {
        float a = 0.f;
        #pragma unroll
        for (int w = 0; w < SW; ++w) {
            int pp = start + w;
            int pcc = pp < (S_LEN - 1) ? pp : (S_LEN - 1);
            a += s[w] * inv * emb[(size_t)pcc * HID + h];
        }
        sv[h] = es[h];
        sv[HID + h] = ee[h];
        sv[2 * HID + h] = a;
    }
}

// ---------------- mention score init ----------------
__global__ void ms_init_kernel(float* __restrict__ msc, const float* __restrict__ b2,
                               int nspans, int npad) {
    int i = blockIdx.x * 256 + threadIdx.x;
    if (i >= npad) return;
    msc[i] = (i < nspans) ? b2[0] : -3e38f;
}

// ---------------- deterministic top-k (single block, iterative argmax) ----------------
__global__ __launch_bounds__(1024) void topk_kernel(float* __restrict__ sc, int n, int k,
                                                    int* __restrict__ idx_out) {
    __shared__ float sv[1024];
    __shared__ int   si[1024];
    int tid = threadIdx.x;
    for (int it = 0; it < k; ++it) {
        float best = -3.3e38f; int bi = 0x7fffffff;
        for (int i = tid; i < n; i += 1024) {
            float v = sc[i];
            if (v > best || (v == best && i < bi)) { best = v; bi = i; }
        }
        sv[tid] = best; si[tid] = bi;
        __syncthreads();
        for (int off = 512; off; off >>= 1) {
            if (tid < off) {
                if (sv[tid + off] > sv[tid] ||
                    (sv[tid + off] == sv[tid] && si[tid + off] < si[tid])) {
                    sv[tid] = sv[tid + off]; si[tid] = si[tid + off];
                }
            }
            __syncthreads();
        }
        if (tid == 0) {
            idx_out[it] = si[0];
            if (si[0] < n) sc[si[0]] = -3.3e38f;
        }
        __syncthreads();
    }
}

// ---------------- s_i / s_j dots over pruned spans ----------------
__global__ __launch_bounds__(256) void sij_kernel(const float* __restrict__ span_vec,
                                                  const int* __restrict__ idx,
                                                  const float* __restrict__ antW,
                                                  float* __restrict__ s_i,
                                                  float* __restrict__ s_j) {
    int r = blockIdx.x;
    int row = idx[r];
    const float* sv = span_vec + (size_t)row * D3;
    float pi = 0.f, pj = 0.f;
    for (int h = threadIdx.x; h < D3; h += 256) {
        float v = sv[h];
        pi += v * antW[h];
        pj += v * antW[D3 + h];
    }
    __shared__ float ri[256], rj[256];
    ri[threadIdx.x] = pi; rj[threadIdx.x] = pj;
    __syncthreads();
    for (int off = 128; off; off >>= 1) {
        if (threadIdx.x < off) {
            ri[threadIdx.x] += ri[threadIdx.x + off];
            rj[threadIdx.x] += rj[threadIdx.x + off];
        }
        __syncthreads();
    }
    if (threadIdx.x == 0) { s_i[r] = ri[0]; s_j[r] = rj[0]; }
}

// ---------------- antecedent scores with dummy column ----------------
__global__ void ant_kernel(const float* __restrict__ s_i, const float* __restrict__ s_j,
                           const float* __restrict__ antb, float* __restrict__ out, int k) {
    int t = blockIdx.x * 256 + threadIdx.x;
    int cols = k + 1;
    if (t >= k * cols) return;
    int i = t / cols, j = t % cols;
    float v;
    if (j == 0) v = 0.f;
    else if (j - 1 >= i) v = -10000.f;
    else v = s_i[i] + s_j[j - 1] + antb[0];
    out[t] = v;
}

// ================= host launcher =================
extern "C" void kernel_launch(void* const* d_in, const int* in_sizes, int n_in,
                              void* d_out, int out_size, void* d_ws, size_t ws_size,
                              hipStream_t stream) {
    const float* x      = (const float*)d_in[0];
    const int*   edges  = (const int*)  d_in[1];
    const float* gW     = (const float*)d_in[2];
    const float* gb     = (const float*)d_in[3];
    const float* attn_w = (const float*)d_in[4];
    const float* attn_b = (const float*)d_in[5];
    const float* W1     = (const float*)d_in[6];
    const float* b1     = (const float*)d_in[7];
    const float* w2     = (const float*)d_in[8];
    const float* b2     = (const float*)d_in[9];
    const float* antW   = (const float*)d_in[10];
    const float* antb   = (const float*)d_in[11];
    const int*   sst    = (const int*)  d_in[12];
    const int*   sen    = (const int*)  d_in[13];

    const int E      = in_sizes[1] / 3;
    const int nspans = in_sizes[12];
    const int npad   = ((nspans + 63) / 64) * 64;

    // recover k from out_size = k*(k+1)
    int k = 1;
    for (int kk = 1; kk < 4096; ++kk) {
        if ((long long)kk * (kk + 1) == (long long)out_size) { k = kk; break; }
    }

    // ---- workspace carve ----
    char* w = (char*)d_ws;
    size_t off = 0;
    auto carve = [&](size_t bytes) { char* p = w + off; off = (off + bytes + 255) & ~(size_t)255; return p; };
    float* cnt  = (float*)carve((size_t)S_LEN * NL * 4);
    float* deg  = (float*)carve((size_t)S_LEN * 4);
    float* gout = (float*)carve((size_t)S_LEN * HID * 4);
    float* emb  = (float*)carve((size_t)S_LEN * HID * 4);
    float* msc  = (float*)carve((size_t)npad * 4);
    int*   tidx = (int*)  carve((size_t)2048 * 4);
    float* s_i  = (float*)carve((size_t)2048 * 4);
    float* s_j  = (float*)carve((size_t)2048 * 4);
    float* span_vec = (float*)carve((size_t)npad * D3 * 4);
    // agg region is last so its size can flex between the two paths
    const size_t agg_small = (size_t)S_LEN * HID * 4;             // 3 MB
    const size_t agg_big   = (size_t)S_LEN * NL * HID * 4;        // 157 MB
    bool big_path = (ws_size > off) && ((ws_size - off) >= agg_big);
    float* agg = (float*)carve(big_path ? agg_big : agg_small);

    // ---- GCN: counts, degree, bias ----
    hipMemsetAsync(cnt, 0, (size_t)S_LEN * NL * 4, stream);
    cnt_count_kernel<<<(E + 255) / 256, 256, 0, stream>>>(edges, cnt, E);
    deg_kernel<<<(S_LEN + 255) / 256, 256, 0, stream>>>(cnt, deg);
    bias_init_kernel<<<(S_LEN * HID + 255) / 256, 256, 0, stream>>>(cnt, gb, gout);

    // ---- GCN aggregation + einsum as WMMA GEMM ----
    if (big_path) {
        // one scatter pass + one GEMM: out += AGG(1024x38400) @ gcn_W_flat(38400x768)
        hipMemsetAsync(agg, 0, agg_big, stream);
        gcn_scatter_all_kernel<<<(E * 16 + 255) / 256, 256, 0, stream>>>(edges, x, agg, E);
        dim3 g3(S_LEN / 64, HID / 64, NL);
        gemm_gcn_big_kernel<<<g3, 128, 0, stream>>>(agg, gW, gout);
    } else {
        dim3 ggrid(S_LEN / 64, HID / 64);
        for (int l = 0; l < NL; ++l) {
            hipMemsetAsync(agg, 0, agg_small, stream);
            gcn_scatter_kernel<<<(E * 16 + 255) / 256, 256, 0, stream>>>(edges, x, agg, l, E);
            gemm_gcn_kernel<<<ggrid, 128, 0, stream>>>(agg, gW + (size_t)l * HID * HID, gout);
        }
    }

    // ---- emb = x + relu(out/deg) ----
    emb_kernel<<<(S_LEN * HID + 255) / 256, 256, 0, stream>>>(x, gout, deg, emb);

    // ---- spans ----
    if (npad > nspans)
        hipMemsetAsync(span_vec + (size_t)nspans * D3, 0,
                       (size_t)(npad - nspans) * D3 * 4, stream);
    span_kernel<<<nspans, 320, 0, stream>>>(emb, sst, sen, attn_w, attn_b, span_vec);

    // ---- mention scores: init + fused WMMA GEMM ----
    ms_init_kernel<<<(npad + 255) / 256, 256, 0, stream>>>(msc, b2, nspans, npad);
    dim3 mgrid(npad / 64, D3 / 64);
    gemm_mention_kernel<<<mgrid, 128, 0, stream>>>(span_vec, W1, b1, w2, msc, nspans);

    // ---- top-k (destructive on msc) ----
    topk_kernel<<<1, 1024, 0, stream>>>(msc, nspans, k, tidx);

    // ---- antecedent scoring ----
    sij_kernel<<<k, 256, 0, stream>>>(span_vec, tidx, antW, s_i, s_j);
    ant_kernel<<<(k * (k + 1) + 255) / 256, 256, 0, stream>>>(s_i, s_j, antb, (float*)d_out, k);
}